// GRNECM_15307263443309
// MI455X (gfx1250) — compile-verified
//
#include <hip/hip_runtime.h>
#include <hip/hip_bf16.h>

#define FEAT   64
#define SPB    8          // segments per block (N divisible: 50000/8 = 6250 blocks)
#define EPB    64         // edges per LDS chunk (64 * 256B = 16KB per buffer)
#define NSEG   50000
#define NEDGE  1600000

typedef unsigned int       u32;
typedef unsigned long long u64;
typedef u32 v4u __attribute__((ext_vector_type(4)));
typedef int v8i __attribute__((ext_vector_type(8)));
typedef int v4i __attribute__((ext_vector_type(4)));

// ---------------------------------------------------------------------------
// Kernel 1: out[n, f] = bias[f]  (covers empty segments; non-empty rows are
// overwritten by the segment kernel). 12.8 MB write, negligible.
// ---------------------------------------------------------------------------
__global__ __launch_bounds__(256) void gat_bias_init(const float* __restrict__ bias,
                                                     float* __restrict__ out) {
    int idx = blockIdx.x * 256 + threadIdx.x;          // grid sized exactly N*F/256
    out[idx] = bias[idx & (FEAT - 1)];
}

// ---------------------------------------------------------------------------
// Kernel 2: weighted segment-sum. Block owns segments [8b, 8b+8); finds its
// edge range by binary search (seg ids are sorted), streams 64-edge chunks of
// neighbor_feats into double-buffered LDS via the Tensor Data Mover
// (tensor_load_to_lds, TENSORcnt), and run-length accumulates per column.
// All stores exclusive -> deterministic, no atomics.
// ---------------------------------------------------------------------------
__global__ __launch_bounds__(64) void gat_segsum(const float* __restrict__ nb,
                                                 const float* __restrict__ attn,
                                                 const int*   __restrict__ seg,
                                                 const float* __restrict__ bias,
                                                 float*       __restrict__ out) {
    __shared__ float fbuf[2][EPB * FEAT];   // 2 x 16KB feature tiles
    __shared__ float abuf[2][EPB];
    __shared__ int   sbuf[2][EPB];
    __shared__ int   erange[2];

    const int t   = threadIdx.x;            // t == feature column owned
    const int sLo = blockIdx.x * SPB;

    if (t < 2) {                            // two parallel lower_bound searches
        int key = sLo + t * SPB;
        int lo = 0, hi = NEDGE;
        while (lo < hi) {
            int mid = (lo + hi) >> 1;
            if (seg[mid] < key) lo = mid + 1; else hi = mid;
        }
        erange[t] = lo;
    }
    __syncthreads();
    const int eLo = erange[0];
    const int eHi = erange[1];
    const int nE  = eHi - eLo;
    if (nE <= 0) return;                    // uniform across block
    const int nch = (nE + EPB - 1) / EPB;

    const float bc = bias[t];

    auto stage = [&](int k) {
        const int b    = k & 1;
        const int base = eLo + k * EPB;
        const int rows = min(EPB, eHi - base);
        if (t < rows) {                     // coalesced attn/seg staging
            abuf[b][t] = attn[base + t];
            sbuf[b][t] = seg[base + t];
        }
        if (t < 32) {                       // wave 0 issues one TDM tile load
            u64 ga = (u64)(uintptr_t)(nb + (u64)base * FEAT);
            u32 la = (u32)(uintptr_t)(&fbuf[b][0]);   // low 32 bits = LDS offset
            u32 td1 = (u32)rows;                      // exact -> no OOB rows
            v4u g0;
            g0[0] = 1u;                               // count=1, user descriptor
            g0[1] = la;                               // lds_addr (bytes)
            g0[2] = (u32)ga;                          // global_addr[31:0]
            g0[3] = ((u32)(ga >> 32) & 0x1FFFFFFu)    // global_addr[56:32]
                    | (2u << 30);                     // type=2 ("image")
            v8i g1;
            g1[0] = (int)(2u << 16);                  // data_size=4B; wg_mask=0
            g1[1] = (int)(64u << 16);                 // tensor_dim0=64 (lo16)
            g1[2] = (int)((td1 & 0xFFFFu) << 16);     // tdim0 hi=0 | tensor_dim1 lo16
            g1[3] = (int)((64u << 16) | (td1 >> 16)); // tensor_dim1 hi16 | tile_dim0=64
            g1[4] = (int)td1;                         // tile_dim1=rows; tile_dim2=0
            g1[5] = 64;                               // tensor_dim0_stride lo32 = 64
            g1[6] = (int)(64u << 16);                 // stride0 hi=0 | tensor_dim1_stride lo16=64
            g1[7] = 0;                                // tensor_dim1_stride hi32
            v4i gz4 = {0, 0, 0, 0};                   // groups 2/3 unused (2D tile)
            v8i gz8 = {0, 0, 0, 0, 0, 0, 0, 0};
            __builtin_amdgcn_tensor_load_to_lds(g0, g1, gz4, gz4, gz8, 0);
        }
    };

    stage(0);

    float acc = 0.0f;
    int   cur = -1;

    for (int k = 0; k < nch; ++k) {
        if (k + 1 < nch) {
            stage(k + 1);                            // prefetch next chunk
            if (t < 32) __builtin_amdgcn_s_wait_tensorcnt(1);  // in-order: chunk k done
        } else {
            if (t < 32) __builtin_amdgcn_s_wait_tensorcnt(0);
        }
        __syncthreads();                             // tile + attn/seg visible to all

        const int b    = k & 1;
        const int rows = min(EPB, eHi - (eLo + k * EPB));
        for (int i = 0; i < rows; ++i) {
            int s = sbuf[b][i];                      // LDS broadcast read
            if (s != cur) {
                if (cur >= 0) out[cur * FEAT + t] = acc + bc;  // exclusive store
                acc = 0.0f;
                cur = s;
            }
            acc = fmaf(fbuf[b][i * FEAT + t], abuf[b][i], acc);  // bank-conflict-free
        }
        __syncthreads();                             // protect buf before restage
    }
    if (cur >= 0) out[cur * FEAT + t] = acc + bc;
}

// ---------------------------------------------------------------------------
extern "C" void kernel_launch(void* const* d_in, const int* in_sizes, int n_in,
                              void* d_out, int out_size, void* d_ws, size_t ws_size,
                              hipStream_t stream) {
    // setup_inputs order: nodes(unused), neighbor_feats, attention, bias, segment_ids
    const float* nb   = (const float*)d_in[1];
    const float* attn = (const float*)d_in[2];
    const float* bias = (const float*)d_in[3];
    const int*   seg  = (const int*)d_in[4];
    float*       out  = (float*)d_out;

    gat_bias_init<<<(NSEG * FEAT) / 256, 256, 0, stream>>>(bias, out);
    gat_segsum<<<NSEG / SPB, 64, 0, stream>>>(nb, attn, seg, bias, out);
}